// CRF_18657337934227
// MI455X (gfx1250) — compile-verified
//
#include <hip/hip_runtime.h>

// CRF forward (log partition) on gfx1250 via exp-domain fp32 WMMA.
//
// B=256, T=1024, NT=96. Per step (natural-log semantics, computed in log2 units):
//   u'[b,i] = log2e*h_t[b,i] + mu[b] + log2( sum_j exp2(u[b,j]-mu[b]) * E[i,j] )
// with u = score*log2e, E = exp(trans) (time-invariant), mu = rowmax(u).
// One wave32 per 16-batch tile, fully independent (no barriers); serial over T.
// GEMM per step: [16,96]x[96,96] = 6 C-tiles x 24 K-steps of v_wmma_f32_16x16x4_f32.
// exp/log use raw v_exp_f32 / v_log_f32 (operands provably normal & positive),
// avoiding OCML's denormal/inf fixup sequences.

#define BATCH   256
#define TMAX    1024
#define NTAGS   96
#define PSTRIDE 98   // even (8B-aligned b64 frag loads), avoids LDS bank conflicts

#define LOG2E 1.4426950408889634f
#define LN2   0.6931471805599453f

typedef __attribute__((ext_vector_type(2))) float v2f;
typedef __attribute__((ext_vector_type(8))) float v8f;

__launch_bounds__(32, 1)
__global__ void crf_fwd_wmma(const float* __restrict__ h,
                             const int*   __restrict__ lengths,
                             const float* __restrict__ trans,
                             float*       __restrict__ out) {
  // E in B-fragment layout: Eb[blk][k][lane] = {E[j,i], E[j+1,i]} with
  // i = 16*blk + (lane%16) (N dim), j = 4*k + 2*(lane/16) (K dim); E[j,i]=exp(trans[i,j]).
  __shared__ v2f Eb[6 * 24 * 32];                       // 36 KB
  __shared__ __align__(16) float pbuf[16 * PSTRIDE];    // 6.1 KB, [batch_local][tag]

  const int lane = threadIdx.x & 31;
  const int nn   = lane & 15;   // N index within tile / lane-in-half
  const int hi   = lane >> 4;   // half-wave select
  const int base = blockIdx.x * 16;

  // ---- one-time: build exp(trans) fragments (single wave; LDS is in-order per wave)
  for (int g = 0; g < 6 * 24; ++g) {
    const int blk = g / 24, k = g - 24 * blk;
    const int i = 16 * blk + nn;
    const int j = 4 * k + 2 * hi;
    v2f e;
    e.x = __builtin_amdgcn_exp2f(trans[i * NTAGS + j] * LOG2E);
    e.y = __builtin_amdgcn_exp2f(trans[i * NTAGS + j + 1] * LOG2E);
    Eb[g * 32 + lane] = e;
  }

  // Per-lane constants. C/D layout: element (m, n) with m = r + 8*hi, n = nn.
  int lenr[8], hbase[8];
  #pragma unroll
  for (int r = 0; r < 8; ++r) {
    const int b = base + r + 8 * hi;
    lenr[r]  = lengths[b];
    hbase[r] = b * (TMAX * NTAGS) + nn;   // + t*NTAGS + 16*blk at use
  }

  // Wave-uniform trip count: once t >= max(lenr) every lane is masked and the
  // recursion is frozen, so stop there (scores beyond are select-preserved).
  int ml = lenr[0];
  #pragma unroll
  for (int r = 1; r < 8; ++r) ml = (lenr[r] > ml) ? lenr[r] : ml;
  #pragma unroll
  for (int d = 1; d <= 16; d <<= 1) {
    const int o = __shfl_xor(ml, d, 32);
    ml = (o > ml) ? o : ml;
  }
  const int tmaxw = __builtin_amdgcn_readfirstlane(ml);

  // ---- init scores (log2 units): -10000*log2e everywhere, 0 at start_tag = 94 = 16*5+14
  float s[6][8];
  #pragma unroll
  for (int blk = 0; blk < 6; ++blk)
    #pragma unroll
    for (int r = 0; r < 8; ++r)
      s[blk][r] = -14426.950408889634f;
  if (nn == 14) {
    #pragma unroll
    for (int r = 0; r < 8; ++r) s[5][r] = 0.0f;
  }

  const int arow = nn * PSTRIDE + 2 * hi;  // A-frag: lane l -> p[l%16, 4k + 2*(l/16) + {0,1}]

  for (int t = 0; t < tmaxw; ++t) {
    // Emissions for this step; issued first, consumed at the very end of the
    // step (hidden under the max/exp/LDS/WMMA work; h is L2-resident).
    float hcur[6][8];
    #pragma unroll
    for (int blk = 0; blk < 6; ++blk)
      #pragma unroll
      for (int r = 0; r < 8; ++r)
        hcur[blk][r] = h[hbase[r] + t * NTAGS + 16 * blk];

    // 1) per-batch rowmax over all 96 tags: reduce over blk (regs) then over n (16 lanes)
    float mv[8];
    #pragma unroll
    for (int r = 0; r < 8; ++r) {
      float m = s[0][r];
      #pragma unroll
      for (int blk = 1; blk < 6; ++blk) m = fmaxf(m, s[blk][r]);
      mv[r] = m;
    }
    #pragma unroll
    for (int d = 1; d <= 8; d <<= 1)
      #pragma unroll
      for (int r = 0; r < 8; ++r)
        mv[r] = fmaxf(mv[r], __shfl_xor(mv[r], d, 32));

    // 2) p = exp2(u - mu) -> LDS in [batch][tag] order (C-layout -> A-layout reshape)
    #pragma unroll
    for (int blk = 0; blk < 6; ++blk)
      #pragma unroll
      for (int r = 0; r < 8; ++r)
        pbuf[(r + 8 * hi) * PSTRIDE + 16 * blk + nn] =
            __builtin_amdgcn_exp2f(s[blk][r] - mv[r]);

    // 3) q = p * E  via fp32 WMMA: 6 output tiles, K=96 in 24 steps of K=4
    v8f acc[6] = {};
    #pragma unroll 4
    for (int k = 0; k < 24; ++k) {
      const v2f a = *(const v2f*)&pbuf[arow + 4 * k];   // A frag shared by all 6 tiles
      #pragma unroll
      for (int blk = 0; blk < 6; ++blk) {
        acc[blk] = __builtin_amdgcn_wmma_f32_16x16x4_f32(
            false, a, false, Eb[(blk * 24 + k) * 32 + lane],
            (short)0, acc[blk], false, false);
      }
    }

    // 4) u' = log2e*h + mu + log2(q), masked by sequence length.
    //    q is always normal & positive (rowmax guarantees one p == 1), so raw
    //    v_log_f32 is exact enough. The empty asm pins ns so the compiler can't
    //    sink the log chain under the mask (select stays a single v_cndmask).
    #pragma unroll
    for (int blk = 0; blk < 6; ++blk)
      #pragma unroll
      for (int r = 0; r < 8; ++r) {
        float ns = fmaf(hcur[blk][r], LOG2E,
                        mv[r] + __builtin_amdgcn_logf(acc[blk][r]));
        asm("" : "+v"(ns));
        s[blk][r] = (t < lenr[r]) ? ns : s[blk][r];
      }
  }

  // ---- epilogue: + trans[end_tag=95, :]*log2e, then logsumexp (log2 domain), x ln2
  #pragma unroll
  for (int blk = 0; blk < 6; ++blk) {
    const float te = trans[(NTAGS - 1) * NTAGS + 16 * blk + nn];
    #pragma unroll
    for (int r = 0; r < 8; ++r) s[blk][r] = fmaf(te, LOG2E, s[blk][r]);
  }
  float mv[8], sm[8];
  #pragma unroll
  for (int r = 0; r < 8; ++r) {
    float m = s[0][r];
    #pragma unroll
    for (int blk = 1; blk < 6; ++blk) m = fmaxf(m, s[blk][r]);
    mv[r] = m;
  }
  #pragma unroll
  for (int d = 1; d <= 8; d <<= 1)
    #pragma unroll
    for (int r = 0; r < 8; ++r)
      mv[r] = fmaxf(mv[r], __shfl_xor(mv[r], d, 32));
  #pragma unroll
  for (int r = 0; r < 8; ++r) {
    float a = 0.0f;
    #pragma unroll
    for (int blk = 0; blk < 6; ++blk)
      a += __builtin_amdgcn_exp2f(s[blk][r] - mv[r]);
    sm[r] = a;
  }
  #pragma unroll
  for (int d = 1; d <= 8; d <<= 1)
    #pragma unroll
    for (int r = 0; r < 8; ++r)
      sm[r] += __shfl_xor(sm[r], d, 32);

  if (nn == 0) {
    #pragma unroll
    for (int r = 0; r < 8; ++r)
      out[base + r + 8 * hi] =
          (mv[r] + __builtin_amdgcn_logf(sm[r])) * LN2;
  }
}

extern "C" void kernel_launch(void* const* d_in, const int* in_sizes, int n_in,
                              void* d_out, int out_size, void* d_ws, size_t ws_size,
                              hipStream_t stream) {
  (void)in_sizes; (void)n_in; (void)out_size; (void)d_ws; (void)ws_size;
  const float* h       = (const float*)d_in[0];
  const int*   lengths = (const int*)d_in[1];
  const float* trans   = (const float*)d_in[2];
  float*       out     = (float*)d_out;
  crf_fwd_wmma<<<dim3(BATCH / 16), dim3(32), 0, stream>>>(h, lengths, trans, out);
}